// GHV_32272384262671
// MI455X (gfx1250) — compile-verified
//
// SC2-PCR-style RANSAC registration for MI455X (gfx1250), wave32.
// CDNA5 paths used: V_WMMA_F32_16X16X4_F32 (K=16 fitness GEMM),
// GLOBAL_LOAD_ASYNC_TO_LDS_B128 + s_wait_asynccnt (LDS staging), global_prefetch_b8.
#include <hip/hip_runtime.h>
#include <hip/hip_bf16.h>
#include <math.h>

#define B_ 8
#define N_ 2048
#define S_ 1024          // N/2 seeds
#define G_ 30
#define T_ITERS_ 10
#define SIGMA_INV 100.0f // 1/0.01
#define THRESH2 0.01f    // 0.1^2

typedef __attribute__((ext_vector_type(2))) float v2f;
typedef __attribute__((ext_vector_type(8))) float v8f;

// ---------------- deterministic counter-hash Gumbel RNG ----------------
__device__ __forceinline__ unsigned mix32(unsigned x) {
  x ^= x >> 16; x *= 0x7feb352dU;
  x ^= x >> 15; x *= 0x846ca68bU;
  x ^= x >> 16; return x;
}
__device__ __forceinline__ float gumbel_rng(unsigned s0, unsigned s1, unsigned s2) {
  unsigned h = mix32(s0 * 0x9E3779B9U ^ mix32(s1 * 0x85EBCA77U + s2));
  float u = ((float)h + 0.5f) * 2.3283064365386963e-10f; // (0,1)
  u = fmaxf(u, 1e-9f);
  return -logf(-logf(u));
}

__device__ __forceinline__ float sanitize_w(float w) {
  if (w != w) w = 1e-7f;       // NaN -> 1e-7
  if (isinf(w)) w = 1.0f;      // +/-Inf -> 1.0
  return w;
}

// CDNA5 async memory->LDS copy (16B per lane), tracked by ASYNCcnt.
// LDS byte offset = low 32 bits of the generic LDS pointer (ISA aperture truncation).
__device__ __forceinline__ void async_copy_b128(const float* lds_dst, const float* gsrc) {
  unsigned lds_off = (unsigned)(unsigned long long)(uintptr_t)lds_dst;
  asm volatile("global_load_async_to_lds_b128 %0, %1, off"
               :: "v"(lds_off), "v"(gsrc) : "memory");
}
__device__ __forceinline__ void wait_asynccnt0() {
  asm volatile("s_wait_asynccnt 0" ::: "memory");
}

// ---------------- init: zero output, dist = 1e8 ----------------
__global__ void k_init(float* out, float* dist) {
  int t = threadIdx.x;
  if (t < B_ * 12) out[t] = 0.0f;   // 8*9 R + 8*3 t = 96
  if (t == 0) dist[0] = 1e8f;
}

// ---------------- softmax over sanitized weights, per batch ----------------
__global__ __launch_bounds__(256) void k_softmax(const float* __restrict__ win,
                                                 float* __restrict__ wsoft) {
  __shared__ float red[256];
  int b = blockIdx.x, t = threadIdx.x;
  float mx = -3.4e38f;
  for (int n = t; n < N_; n += 256) mx = fmaxf(mx, sanitize_w(win[b * N_ + n]));
  red[t] = mx; __syncthreads();
  for (int off = 128; off > 0; off >>= 1) { if (t < off) red[t] = fmaxf(red[t], red[t + off]); __syncthreads(); }
  mx = red[0]; __syncthreads();
  float sum = 0.0f;
  for (int n = t; n < N_; n += 256) sum += expf(sanitize_w(win[b * N_ + n]) - mx);
  red[t] = sum; __syncthreads();
  for (int off = 128; off > 0; off >>= 1) { if (t < off) red[t] += red[t + off]; __syncthreads(); }
  sum = red[0];
  for (int n = t; n < N_; n += 256)
    wsoft[b * N_ + n] = expf(sanitize_w(win[b * N_ + n]) - mx) / sum;
}

// ---------------- B-matrix for the K=16 fitness GEMM (iteration-invariant) ----------------
__global__ __launch_bounds__(256) void k_bmat(const float* __restrict__ src,
                                              const float* __restrict__ tgt,
                                              float* __restrict__ Bm,
                                              float* __restrict__ pc2) {
  int idx = blockIdx.x * 256 + threadIdx.x;
  if (idx >= B_ * N_) return;
  int b = idx / N_, n = idx % N_;
  float p0 = src[(b * 3 + 0) * N_ + n], p1 = src[(b * 3 + 1) * N_ + n], p2 = src[(b * 3 + 2) * N_ + n];
  float c0 = tgt[(b * 3 + 0) * N_ + n], c1 = tgt[(b * 3 + 1) * N_ + n], c2 = tgt[(b * 3 + 2) * N_ + n];
  float* Bb = Bm + (size_t)b * 16 * N_;
  Bb[0 * N_ + n] = c0; Bb[1 * N_ + n] = c1; Bb[2 * N_ + n] = c2;
  Bb[3 * N_ + n] = p0; Bb[4 * N_ + n] = p1; Bb[5 * N_ + n] = p2;
  float cs[3] = {c0, c1, c2}, ps[3] = {p0, p1, p2};
#pragma unroll
  for (int i = 0; i < 3; i++)
#pragma unroll
    for (int j = 0; j < 3; j++) Bb[(6 + i * 3 + j) * N_ + n] = cs[i] * ps[j];
  Bb[15 * N_ + n] = 1.0f;
  pc2[b * N_ + n] = p0 * p0 + p1 * p1 + p2 * p2 + c0 * c0 + c1 * c1 + c2 * c2;
}

// ---------------- Gumbel top-1024 seed sampling via stable ranking ----------------
__global__ __launch_bounds__(256) void k_seed_rank(const float* __restrict__ src,
                                                   const float* __restrict__ tgt,
                                                   const float* __restrict__ wsoft,
                                                   float* __restrict__ ssrc,
                                                   float* __restrict__ scorr,
                                                   float* __restrict__ sw, int iter) {
  __shared__ float key[N_];
  int b = blockIdx.x, t = threadIdx.x;
  for (int n = t; n < N_; n += 256) {
    float w = wsoft[b * N_ + n];
    key[n] = logf(w) + gumbel_rng(0x1F123BB5u + (unsigned)iter, (unsigned)b, (unsigned)n);
  }
  __syncthreads();
  for (int n = t; n < N_; n += 256) {
    float kn = key[n];
    int r = 0;
    for (int m = 0; m < N_; ++m) {
      float km = key[m];
      r += (km > kn) || (km == kn && m < n);   // stable descending rank
    }
    if (r < S_) {
      for (int d = 0; d < 3; ++d) {
        ssrc[(b * 3 + d) * S_ + r] = src[(b * 3 + d) * N_ + n];
        scorr[(b * 3 + d) * S_ + r] = tgt[(b * 3 + d) * N_ + n];
      }
      sw[b * S_ + r] = wsoft[b * N_ + n];
    }
  }
}

// ---------------- 3x3 Kabsch rotation: R = V diag(1,1,det) U^T from H = U S V^T ----------------
__device__ void kabsch_rot(const float H[3][3], float R[3][3]) {
  float A[3][3], V[3][3];
  for (int i = 0; i < 3; i++)
    for (int j = 0; j < 3; j++) {
      float s = 0.0f;
      for (int k = 0; k < 3; k++) s += H[k][i] * H[k][j];
      A[i][j] = s;
      V[i][j] = (i == j) ? 1.0f : 0.0f;
    }
  for (int sweep = 0; sweep < 10; ++sweep) {
    for (int p = 0; p < 2; p++)
      for (int q = p + 1; q < 3; q++) {
        float apq = A[p][q];
        if (fabsf(apq) < 1e-14f) continue;
        float tau = (A[q][q] - A[p][p]) / (2.0f * apq);
        float tt = ((tau >= 0.0f) ? 1.0f : -1.0f) / (fabsf(tau) + sqrtf(1.0f + tau * tau));
        float cc = 1.0f / sqrtf(1.0f + tt * tt);
        float ss = tt * cc;
        for (int k = 0; k < 3; k++) { float a0 = A[k][p], a1 = A[k][q]; A[k][p] = cc * a0 - ss * a1; A[k][q] = ss * a0 + cc * a1; }
        for (int k = 0; k < 3; k++) { float a0 = A[p][k], a1 = A[q][k]; A[p][k] = cc * a0 - ss * a1; A[q][k] = ss * a0 + cc * a1; }
        for (int k = 0; k < 3; k++) { float v0 = V[k][p], v1 = V[k][q]; V[k][p] = cc * v0 - ss * v1; V[k][q] = ss * v0 + cc * v1; }
      }
  }
  float ev[3] = {A[0][0], A[1][1], A[2][2]};
  int ord[3] = {0, 1, 2};
  for (int i = 0; i < 2; i++)
    for (int j = 0; j < 2 - i; j++)
      if (ev[ord[j]] < ev[ord[j + 1]]) { int tm = ord[j]; ord[j] = ord[j + 1]; ord[j + 1] = tm; }
  float Vc[3][3];
  for (int k = 0; k < 3; k++)
    for (int i = 0; i < 3; i++) Vc[i][k] = V[i][ord[k]];
  float U[3][3];
  {
    float u[3];
    for (int i = 0; i < 3; i++) u[i] = H[i][0] * Vc[0][0] + H[i][1] * Vc[1][0] + H[i][2] * Vc[2][0];
    float nn = sqrtf(u[0] * u[0] + u[1] * u[1] + u[2] * u[2]);
    if (nn > 1e-12f) { U[0][0] = u[0] / nn; U[1][0] = u[1] / nn; U[2][0] = u[2] / nn; }
    else { U[0][0] = 1.0f; U[1][0] = 0.0f; U[2][0] = 0.0f; }
  }
  {
    float u[3];
    for (int i = 0; i < 3; i++) u[i] = H[i][0] * Vc[0][1] + H[i][1] * Vc[1][1] + H[i][2] * Vc[2][1];
    float d = u[0] * U[0][0] + u[1] * U[1][0] + u[2] * U[2][0];
    for (int i = 0; i < 3; i++) u[i] -= d * U[i][0];
    float nn = sqrtf(u[0] * u[0] + u[1] * u[1] + u[2] * u[2]);
    if (nn > 1e-12f) { U[0][1] = u[0] / nn; U[1][1] = u[1] / nn; U[2][1] = u[2] / nn; }
    else {
      float ax = fabsf(U[0][0]), ay = fabsf(U[1][0]), az = fabsf(U[2][0]);
      float e[3] = {0.0f, 0.0f, 0.0f};
      if (ax <= ay && ax <= az) e[0] = 1.0f; else if (ay <= az) e[1] = 1.0f; else e[2] = 1.0f;
      float cx = U[1][0] * e[2] - U[2][0] * e[1];
      float cy = U[2][0] * e[0] - U[0][0] * e[2];
      float cz = U[0][0] * e[1] - U[1][0] * e[0];
      float cn = fmaxf(sqrtf(cx * cx + cy * cy + cz * cz), 1e-20f);
      U[0][1] = cx / cn; U[1][1] = cy / cn; U[2][1] = cz / cn;
    }
  }
  U[0][2] = U[1][0] * U[2][1] - U[2][0] * U[1][1];
  U[1][2] = U[2][0] * U[0][1] - U[0][0] * U[2][1];
  U[2][2] = U[0][0] * U[1][1] - U[1][0] * U[0][1];
  float detV = Vc[0][0] * (Vc[1][1] * Vc[2][2] - Vc[1][2] * Vc[2][1])
             - Vc[0][1] * (Vc[1][0] * Vc[2][2] - Vc[1][2] * Vc[2][0])
             + Vc[0][2] * (Vc[1][0] * Vc[2][1] - Vc[1][1] * Vc[2][0]);
  float d3 = (detV >= 0.0f) ? 1.0f : -1.0f;   // det(U)=+1 by construction
  for (int i = 0; i < 3; i++)
    for (int j = 0; j < 3; j++)
      R[i][j] = Vc[i][0] * U[j][0] + Vc[i][1] * U[j][1] + d3 * Vc[i][2] * U[j][2];
}

// ---------------- per-(b,s): async LDS stage -> consensus row -> Gumbel top-30 -> Kabsch ----------------
__global__ __launch_bounds__(256) void k_consensus(const float* __restrict__ ssrc,
                                                   const float* __restrict__ scorr,
                                                   const float* __restrict__ sw,
                                                   float* __restrict__ Rws,
                                                   float* __restrict__ tws,
                                                   float* __restrict__ A16,
                                                   int* __restrict__ counts, int iter) {
  __shared__ __attribute__((aligned(16))) float lsx[S_], lsy[S_], lsz[S_];
  __shared__ __attribute__((aligned(16))) float lcx[S_], lcy[S_], lcz[S_];
  __shared__ __attribute__((aligned(16))) float lw[S_];
  __shared__ float key[S_];
  __shared__ float redf[256];
  __shared__ float wbv[8];
  __shared__ int wbi[8];
  __shared__ int sel[G_];
  int b = blockIdx.x >> 10;
  int s = blockIdx.x & (S_ - 1);
  int t = threadIdx.x;
  int wv = t >> 5, ln = t & 31;

  // Async memory->LDS staging: 16B per lane per array (256 lanes cover 1024 floats).
  {
    int j0 = t * 4;
    async_copy_b128(&lsx[j0], ssrc + (size_t)(b * 3 + 0) * S_ + j0);
    async_copy_b128(&lsy[j0], ssrc + (size_t)(b * 3 + 1) * S_ + j0);
    async_copy_b128(&lsz[j0], ssrc + (size_t)(b * 3 + 2) * S_ + j0);
    async_copy_b128(&lcx[j0], scorr + (size_t)(b * 3 + 0) * S_ + j0);
    async_copy_b128(&lcy[j0], scorr + (size_t)(b * 3 + 1) * S_ + j0);
    async_copy_b128(&lcz[j0], scorr + (size_t)(b * 3 + 2) * S_ + j0);
    async_copy_b128(&lw[j0], sw + (size_t)b * S_ + j0);
    wait_asynccnt0();
  }
  __syncthreads();

  float px = lsx[s], py = lsy[s], pz = lsz[s];
  float qx = lcx[s], qy = lcy[s], qz = lcz[s];
  float part = 0.0f;
  for (int j = t; j < S_; j += 256) {
    float dx = lsx[j] - px, dy = lsy[j] - py, dz = lsz[j] - pz;
    float ds = sqrtf(dx * dx + dy * dy + dz * dz);
    float ex = lcx[j] - qx, ey = lcy[j] - qy, ez = lcz[j] - qz;
    float dc = sqrtf(ex * ex + ey * ey + ez * ez);
    float df = ds - dc;
    float cons = fmaxf(expf(-df * df * SIGMA_INV), 0.1f);
    key[j] = cons;
    part += cons;
  }
  redf[t] = part; __syncthreads();
  for (int off = 128; off > 0; off >>= 1) { if (t < off) redf[t] += redf[t + off]; __syncthreads(); }
  float logsum = logf(redf[0]);
  __syncthreads();
  unsigned rowid = (unsigned)blockIdx.x;
  for (int j = t; j < S_; j += 256)
    key[j] = logf(key[j]) - logsum + gumbel_rng(0xB5297A4Du + (unsigned)iter, rowid, (unsigned)j);
  __syncthreads();

  // 30x stable argmax: wave32 shfl reduce + 8-wave combine (2 barriers/pass)
  for (int g = 0; g < G_; ++g) {
    float bv = -3.4e38f; int bi = S_;
    for (int j = t; j < S_; j += 256) {
      float v = key[j];
      if (v > bv || (v == bv && j < bi)) { bv = v; bi = j; }
    }
    for (int off = 16; off > 0; off >>= 1) {
      float ov = __shfl_down(bv, off);
      int oi = __shfl_down(bi, off);
      if (ov > bv || (ov == bv && oi < bi)) { bv = ov; bi = oi; }
    }
    if (ln == 0) { wbv[wv] = bv; wbi[wv] = bi; }
    __syncthreads();
    if (t == 0) {
      float fb = wbv[0]; int fi = wbi[0];
      for (int w2 = 1; w2 < 8; w2++)
        if (wbv[w2] > fb || (wbv[w2] == fb && wbi[w2] < fi)) { fb = wbv[w2]; fi = wbi[w2]; }
      sel[g] = fi;
      key[fi] = -3.4e38f;
    }
    __syncthreads();
  }

  if (t == 0) {
    float wsum = 0.0f, msx = 0, msy = 0, msz = 0, mcx = 0, mcy = 0, mcz = 0;
    for (int g = 0; g < G_; ++g) {
      int j = sel[g];
      wsum += lw[j];
      msx += lsx[j]; msy += lsy[j]; msz += lsz[j];
      mcx += lcx[j]; mcy += lcy[j]; mcz += lcz[j];
    }
    const float inv = 1.0f / (float)G_;
    msx *= inv; msy *= inv; msz *= inv; mcx *= inv; mcy *= inv; mcz *= inv;
    wsum = fmaxf(wsum, 1e-30f);
    float H[3][3] = {{0, 0, 0}, {0, 0, 0}, {0, 0, 0}};
    for (int g = 0; g < G_; ++g) {
      int j = sel[g];
      float wg = lw[j] / wsum;
      float ax = lsx[j] - msx, ay = lsy[j] - msy, az = lsz[j] - msz;
      float bx = lcx[j] - mcx, by = lcy[j] - mcy, bz = lcz[j] - mcz;
      H[0][0] += wg * ax * bx; H[0][1] += wg * ax * by; H[0][2] += wg * ax * bz;
      H[1][0] += wg * ay * bx; H[1][1] += wg * ay * by; H[1][2] += wg * ay * bz;
      H[2][0] += wg * az * bx; H[2][1] += wg * az * by; H[2][2] += wg * az * bz;
    }
    float R[3][3];
    kabsch_rot(H, R);
    float t0 = -(R[0][0] * msx + R[0][1] * msy + R[0][2] * msz) + mcx;
    float t1 = -(R[1][0] * msx + R[1][1] * msy + R[1][2] * msz) + mcy;
    float t2 = -(R[2][0] * msx + R[2][1] * msy + R[2][2] * msz) + mcz;
    size_t base = (size_t)(b * S_ + s);
    float* Rp = Rws + base * 9;
    for (int i = 0; i < 3; i++)
      for (int j = 0; j < 3; j++) Rp[i * 3 + j] = R[i][j];
    float* Tp = tws + base * 3;
    Tp[0] = t0; Tp[1] = t1; Tp[2] = t2;
    float q0 = R[0][0] * t0 + R[1][0] * t1 + R[2][0] * t2;   // q = R^T t
    float q1 = R[0][1] * t0 + R[1][1] * t1 + R[2][1] * t2;
    float q2 = R[0][2] * t0 + R[1][2] * t1 + R[2][2] * t2;
    float* Ap = A16 + base * 16;
    Ap[0] = -2.0f * t0; Ap[1] = -2.0f * t1; Ap[2] = -2.0f * t2;
    Ap[3] = 2.0f * q0;  Ap[4] = 2.0f * q1;  Ap[5] = 2.0f * q2;
    for (int i = 0; i < 3; i++)
      for (int j = 0; j < 3; j++) Ap[6 + i * 3 + j] = -2.0f * R[i][j];
    Ap[15] = t0 * t0 + t1 * t1 + t2 * t2;
    counts[base] = 0;
  }
}

// ---------------- fitness: [S x N] = A[S x 16] * B[16 x N] via V_WMMA_F32_16X16X4_F32 ----------------
__global__ __launch_bounds__(256) void k_fitness(const float* __restrict__ A16,
                                                 const float* __restrict__ Bm,
                                                 const float* __restrict__ pc2,
                                                 int* __restrict__ counts) {
  int b = blockIdx.x >> 6;
  int stile = blockIdx.x & 63;
  int wave = threadIdx.x >> 5;
  int lane = threadIdx.x & 31;
  int half = lane >> 4;      // 0: lanes 0-15, 1: lanes 16-31
  int lan = lane & 15;       // A row (M) / B column within tile
  int sbase = stile * 16;
  // A fragments (ISA 16x4 f32 layout: lanes<16 -> K 0,1; lanes>=16 -> K 2,3)
  v2f a[4];
  const float* Arow = A16 + (size_t)(b * S_ + sbase + lan) * 16;
#pragma unroll
  for (int kk = 0; kk < 4; kk++) {
    a[kk].x = Arow[4 * kk + 2 * half + 0];
    a[kk].y = Arow[4 * kk + 2 * half + 1];
  }
  int cnt[8];
#pragma unroll
  for (int r = 0; r < 8; r++) cnt[r] = 0;
  int nstart = wave * 256;   // each of 8 waves handles a 256-column strip
  for (int nt = 0; nt < 16; ++nt) {
    int n = nstart + nt * 16 + lan;
    if (nt + 1 < 16)   // prefetch next B column group (lowers to global_prefetch_b8)
      __builtin_prefetch(&Bm[(size_t)(b * 16) * N_ + n + 16], 0, 1);
    v2f bf[4];
#pragma unroll
    for (int kk = 0; kk < 4; kk++) {   // B 4x16: row striping mirrors A
      bf[kk].x = Bm[(size_t)(b * 16 + 4 * kk + 2 * half + 0) * N_ + n];
      bf[kk].y = Bm[(size_t)(b * 16 + 4 * kk + 2 * half + 1) * N_ + n];
    }
    v8f c = {0.f, 0.f, 0.f, 0.f, 0.f, 0.f, 0.f, 0.f};
#pragma unroll
    for (int kk = 0; kk < 4; kk++)
      c = __builtin_amdgcn_wmma_f32_16x16x4_f32(false, a[kk], false, bf[kk],
                                                (short)0, c, false, false);
    float pcv = pc2[b * N_ + n];
#pragma unroll
    for (int r = 0; r < 8; r++) cnt[r] += (c[r] + pcv < THRESH2) ? 1 : 0;
  }
  // reduce over the 16 columns held by each half-wave; D layout: M = r + 8*half
#pragma unroll
  for (int r = 0; r < 8; r++) {
    int v = cnt[r];
    for (int m = 1; m < 16; m <<= 1) v += __shfl_xor(v, m, 16);
    if (lan == 0) atomicAdd(&counts[b * S_ + sbase + 8 * half + r], v);
  }
}

// ---------------- best-of selection: vv < dist, copy winning R/t ----------------
__global__ __launch_bounds__(256) void k_select(const int* __restrict__ counts,
                                                const float* __restrict__ Rws,
                                                const float* __restrict__ tws,
                                                float* __restrict__ dist,
                                                float* __restrict__ out) {
  __shared__ int sbest[B_];
  __shared__ int smax[B_];
  __shared__ int flag;
  int t = threadIdx.x;
  int grp = t >> 5, lane = t & 31;
  if (grp < B_) {
    int bm = -1, bi = S_;
    for (int s = lane; s < S_; s += 32) {
      int c = counts[grp * S_ + s];
      if (c > bm || (c == bm && s < bi)) { bm = c; bi = s; }
    }
    for (int off = 16; off > 0; off >>= 1) {
      int om = __shfl_down(bm, off);
      int oi = __shfl_down(bi, off);
      if (om > bm || (om == bm && oi < bi)) { bm = om; bi = oi; }
    }
    if (lane == 0) { smax[grp] = bm; sbest[grp] = bi; }
  }
  __syncthreads();
  if (t == 0) {
    float vv = 0.0f;
    for (int b = 0; b < B_; b++) vv += (float)smax[b];
    vv /= (float)(B_ * N_);
    flag = (vv < dist[0]) ? 1 : 0;
    if (flag) dist[0] = vv;
  }
  __syncthreads();
  if (flag) {
    if (t < B_ * 9) {
      int b = t / 9, e = t % 9;
      out[t] = Rws[((size_t)(b * S_ + sbest[b])) * 9 + e];
    } else if (t < B_ * 12) {
      int u = t - B_ * 9;
      int b = u / 3, e = u % 3;
      out[B_ * 9 + u] = tws[((size_t)(b * S_ + sbest[b])) * 3 + e];
    }
  }
}

// ---------------- host-side orchestration ----------------
extern "C" void kernel_launch(void* const* d_in, const int* in_sizes, int n_in,
                              void* d_out, int out_size, void* d_ws, size_t ws_size,
                              hipStream_t stream) {
  (void)in_sizes; (void)n_in; (void)out_size; (void)ws_size;
  const float* src = (const float*)d_in[0];   // [B,3,N]
  const float* tgt = (const float*)d_in[1];   // [B,3,N]
  const float* win = (const float*)d_in[2];   // [B,N]
  float* out = (float*)d_out;                 // 72 R + 24 t

  char* ws = (char*)d_ws;
  size_t off = 0;
  auto alloc = [&](size_t bytes) -> void* {
    off = (off + 255) & ~(size_t)255;
    void* p = ws + off;
    off += bytes;
    return p;
  };
  float* wsoft = (float*)alloc((size_t)B_ * N_ * 4);
  float* ssrc  = (float*)alloc((size_t)B_ * 3 * S_ * 4);
  float* scorr = (float*)alloc((size_t)B_ * 3 * S_ * 4);
  float* sw    = (float*)alloc((size_t)B_ * S_ * 4);
  float* Rws   = (float*)alloc((size_t)B_ * S_ * 9 * 4);
  float* tws   = (float*)alloc((size_t)B_ * S_ * 3 * 4);
  float* A16   = (float*)alloc((size_t)B_ * S_ * 16 * 4);
  float* Bm    = (float*)alloc((size_t)B_ * 16 * N_ * 4);
  float* pc2   = (float*)alloc((size_t)B_ * N_ * 4);
  int*   cnts  = (int*)  alloc((size_t)B_ * S_ * 4);
  float* dist  = (float*)alloc(256);

  k_init<<<1, 128, 0, stream>>>(out, dist);
  k_softmax<<<B_, 256, 0, stream>>>(win, wsoft);
  k_bmat<<<(B_ * N_ + 255) / 256, 256, 0, stream>>>(src, tgt, Bm, pc2);
  for (int it = 0; it < T_ITERS_; ++it) {
    k_seed_rank<<<B_, 256, 0, stream>>>(src, tgt, wsoft, ssrc, scorr, sw, it);
    k_consensus<<<B_ * S_, 256, 0, stream>>>(ssrc, scorr, sw, Rws, tws, A16, cnts, it);
    k_fitness<<<B_ * (S_ / 16), 256, 0, stream>>>(A16, Bm, pc2, cnts);
    k_select<<<1, 256, 0, stream>>>(cnts, Rws, tws, dist, out);
  }
}